// RegionLoss_86191403696804
// MI455X (gfx1250) — compile-verified
//
#include <hip/hip_runtime.h>
#include <cstdint>

// ---- problem constants (match reference) ----
#define NB     32
#define NA     5
#define NH     38
#define NW     38
#define NC     80
#define MAXGT  50
#define HW     (NH * NW)            // 1444
#define CB     (NA * HW)            // 7220 cells per batch
#define BLK    256
#define BPB    ((CB + BLK - 1) / BLK)   // 29 blocks per batch
#define NBLOCKS (NB * BPB)              // 928
#define REC    12                       // dwords per GT record

__device__ __constant__ float c_aw[NA] = {1.3221f, 3.19275f, 5.05587f, 9.47112f, 11.2364f};
__device__ __constant__ float c_ah[NA] = {1.73145f, 4.00944f, 8.09892f, 4.84053f, 10.0071f};

typedef __attribute__((ext_vector_type(2))) float v2f;
typedef __attribute__((ext_vector_type(8))) float v8f;

// ---------------------------------------------------------------------------
// Kernel 0: per-GT records. (b,t) -> {gx,gy,gw,gh, tx,ty,tw,th, bn,gi,gj,gcls}
// ---------------------------------------------------------------------------
__global__ void region_prep(const float* __restrict__ target,
                            float* __restrict__ recs) {
    int b = blockIdx.x;
    int t = threadIdx.x;
    if (t >= MAXGT) return;
    const float* tp = target + (size_t)b * (MAXGT * 5) + t * 5;
    float t0 = tp[0], t1 = tp[1], t2 = tp[2], t3 = tp[3], t4 = tp[4];

    int gcls = (int)floorf(t0 * (float)NC);
    gcls = min(max(gcls, 0), NC - 1);
    float gx = t1 * (float)NW;
    float gy = t2 * (float)NH;
    float gw = fmaxf(t3 * (float)NW, 1e-6f);
    float gh = fmaxf(t4 * (float)NH, 1e-6f);

    // best anchor by whole-box IoU (centered at origin) -- first max wins
    int   bn   = 0;
    float best = -1e30f;
    #pragma unroll
    for (int n = 0; n < NA; ++n) {
        float aw = c_aw[n], ah = c_ah[n];
        float inter = fminf(gw, aw) * fminf(gh, ah);
        float uni   = gw * gh + aw * ah - inter;
        float r     = inter / fmaxf(uni, 1e-9f);
        if (r > best) { best = r; bn = n; }
    }
    int gi = min(max((int)floorf(gx), 0), NW - 1);
    int gj = min(max((int)floorf(gy), 0), NH - 1);

    float* r = recs + ((size_t)b * MAXGT + t) * REC;
    r[0] = gx; r[1] = gy; r[2] = gw; r[3] = gh;
    r[4] = gx - (float)gi;
    r[5] = gy - (float)gj;
    r[6] = logf(gw / c_aw[bn]);
    r[7] = logf(gh / c_ah[bn]);
    ((int*)r)[8]  = bn;
    ((int*)r)[9]  = gi;
    ((int*)r)[10] = gj;
    ((int*)r)[11] = gcls;
}

// ---------------------------------------------------------------------------
// Kernel 1: main streaming pass. One block handles 256 cells of one batch.
// GT records are staged into LDS with CDNA5 async global->LDS loads.
// ---------------------------------------------------------------------------
__global__ __launch_bounds__(BLK) void region_main(const float* __restrict__ out,
                                                   const float* __restrict__ recs,
                                                   float* __restrict__ partials) {
    __shared__ float sGT[MAXGT * REC];   // 600 dwords
    __shared__ float sred[BLK];

    int b   = blockIdx.x / BPB;
    int blk = blockIdx.x % BPB;
    int tid = threadIdx.x;

    // ---- async stage of 50 GT records (2400 B) into LDS: 150 x b128 ----
    const float* gsrc = recs + (size_t)b * (MAXGT * REC);
    if (tid < (MAXGT * REC) / 4) {
        unsigned ldsOff = (unsigned)(uintptr_t)(&sGT[0]) + (unsigned)tid * 16u;
        const float* g  = gsrc + tid * 4;
        asm volatile("global_load_async_to_lds_b128 %0, %1, off"
                     :: "v"(ldsOff), "v"(g)
                     : "memory");
    }
    __builtin_amdgcn_s_wait_asynccnt(0);
    __syncthreads();

    float loss = 0.0f;
    int cell = blk * BLK + tid;
    if (cell < CB) {
        int a   = cell / HW;
        int rem = cell - a * HW;
        int j   = rem / NW;
        int i   = rem - j * NW;

        const float* op = out + ((size_t)(b * NA + a) * (5 + NC)) * HW + rem;
        float o0 = op[0];
        float o1 = op[HW];
        float o2 = op[2 * HW];
        float o3 = op[3 * HW];
        float o4 = op[4 * HW];

        float x    = 1.0f / (1.0f + expf(-o0));
        float y    = 1.0f / (1.0f + expf(-o1));
        float w    = o2;
        float h    = o3;
        float conf = 1.0f / (1.0f + expf(-o4));

        float aw = c_aw[a], ah = c_ah[a];
        float px = x + (float)i;
        float py = y + (float)j;
        float pw = expf(w) * aw;
        float ph = expf(h) * ah;

        float best_iou = 0.0f;       // IoUs are >= 0
        int   matched = 0, mcls = 0;
        float mtx = 0, mty = 0, mtw = 0, mth = 0, mtconf = 0;

        for (int t = 0; t < MAXGT; ++t) {
            const float* r = &sGT[t * REC];   // broadcast read, no bank conflict
            float gx = r[0], gy = r[1], gw = r[2], gh = r[3];
            float mx = fminf(px - pw * 0.5f, gx - gw * 0.5f);
            float Mx = fmaxf(px + pw * 0.5f, gx + gw * 0.5f);
            float my = fminf(py - ph * 0.5f, gy - gh * 0.5f);
            float My = fmaxf(py + ph * 0.5f, gy + gh * 0.5f);
            float cw = pw + gw - (Mx - mx);
            float ch = ph + gh - (My - my);
            float inter = fmaxf(cw, 0.0f) * fmaxf(ch, 0.0f);
            float uni   = pw * ph + gw * gh - inter;
            float iou   = inter / fmaxf(uni, 1e-9f);
            best_iou = fmaxf(best_iou, iou);

            int bn = __float_as_int(r[8]);
            int gi = __float_as_int(r[9]);
            int gj = __float_as_int(r[10]);
            if (bn == a && gi == i && gj == j) {   // last GT wins (scatter-set order)
                matched = 1;
                mtx = r[4]; mty = r[5]; mtw = r[6]; mth = r[7];
                mtconf = iou;
                mcls = __float_as_int(r[11]);
            }
        }

        float conf_mask = matched ? 5.0f : (best_iou > 0.6f ? 0.0f : 1.0f);
        float dconf = conf - (matched ? mtconf : 0.0f);
        loss = 0.5f * conf_mask * dconf * dconf;

        if (matched) {
            float dx = x - mtx, dy = y - mty, dw = w - mtw, dh = h - mth;
            loss += 0.5f * (dx * dx + dy * dy + dw * dw + dh * dh);
            // cross-entropy only at matched cells (<=50 per batch)
            const float* cp = op + 5 * HW;
            float mxl = -1e30f;
            for (int c = 0; c < NC; ++c) mxl = fmaxf(mxl, cp[c * HW]);
            float s = 0.0f;
            for (int c = 0; c < NC; ++c) s += expf(cp[c * HW] - mxl);
            loss += (logf(s) + mxl) - cp[mcls * HW];
        }
    }

    // deterministic block reduce
    sred[tid] = loss;
    __syncthreads();
    for (int off = BLK / 2; off > 0; off >>= 1) {
        if (tid < off) sred[tid] += sred[tid + off];
        __syncthreads();
    }
    if (tid == 0) partials[blockIdx.x] = sred[0];
}

// ---------------------------------------------------------------------------
// Kernel 2: single-wave final reduction. Cross-lane sum via WMMA f32 16x16x4:
// A vgpr0 holds lane partials -> A[m][0]=p_m (lanes 0-15), A[m][2]=p_{m+16};
// B = ones => D[m][n] = p_m + p_{m+16}; per-lane sum of 8 D regs gives the
// low/high half-row-group sums; shfl_xor(16) combines them.
// ---------------------------------------------------------------------------
__global__ void region_reduce(const float* __restrict__ partials,
                              float* __restrict__ outp, int n) {
    int lane = threadIdx.x;          // exactly 32 threads, EXEC all-ones
    float p = 0.0f;
    for (int idx = lane; idx < n; idx += 32) p += partials[idx];

    v2f a = {p, 0.0f};
    v2f bv = {1.0f, 1.0f};
    v8f c = {};
    v8f d = __builtin_amdgcn_wmma_f32_16x16x4_f32(
        /*neg_a=*/false, a, /*neg_b=*/false, bv,
        /*c_mod=*/(short)0, c, /*reuse_a=*/false, /*reuse_b=*/false);

    float s = d[0] + d[1] + d[2] + d[3] + d[4] + d[5] + d[6] + d[7];
    s += __shfl_xor(s, 16, 32);
    if (lane == 0) outp[0] = s;
}

// ---------------------------------------------------------------------------
extern "C" void kernel_launch(void* const* d_in, const int* in_sizes, int n_in,
                              void* d_out, int out_size, void* d_ws, size_t ws_size,
                              hipStream_t stream) {
    const float* output = (const float*)d_in[0];   // (32, 425, 38, 38) f32
    const float* target = (const float*)d_in[1];   // (32, 250) f32

    float* recs     = (float*)d_ws;                                        // 76800 B
    float* partials = (float*)((char*)d_ws + (size_t)NB * MAXGT * REC * 4); // 3712 B

    region_prep<<<NB, 64, 0, stream>>>(target, recs);
    region_main<<<NBLOCKS, BLK, 0, stream>>>(output, recs, partials);
    region_reduce<<<1, 32, 0, stream>>>(partials, (float*)d_out, NBLOCKS);
}